// TextGuidedCrossAttention_25933012533624
// MI455X (gfx1250) — compile-verified
//
#include <hip/hip_runtime.h>

// TextGuidedCrossAttention, collapsed via the rank-1 structure of FT = GvT ⊗ l.
//   energy row i = a_i * l_j + const_i  ->  softmax depends only on a_i * l_j
//   out[b,c,i]   = img + gamma * (vg[b,c] * s[b,i] + bv[c])
// with a[b,i] = x_i . (Wq^T (Wk GvT)) + bq.(Wk GvT),  s_i = sum_j l_j softmax_j(a_i l_j).
// bk cancels entirely. HBM traffic ~57MB (~2.4us at 23.3TB/s) + 42M v_exp_f32.
// Text MLP runs on V_WMMA_F32_16X16X4_F32; l is staged to LDS with
// GLOBAL_LOAD_ASYNC_TO_LDS_B32 (ASYNCcnt path).

#define B_   8
#define C_   256
#define C2_  512
#define DK_  32
#define HW_  2304
#define LN_EPS 1e-5f

typedef __attribute__((ext_vector_type(2))) float v2f;
typedef __attribute__((ext_vector_type(8))) float v8f;

// ---------------------------------------------------------------------------
// Kernel 1: text MLP (WMMA f32 16x16x4), then tiny matvecs -> vg, weff, cq.
// Single 256-thread (8-wave) block; total work ~2 MFLOP.
// ---------------------------------------------------------------------------
__global__ __launch_bounds__(256) void tgca_mlp_wmma(
    const float* __restrict__ text, const float* __restrict__ Wg1,
    const float* __restrict__ bg1,  const float* __restrict__ ln_g,
    const float* __restrict__ ln_b, const float* __restrict__ Wg2,
    const float* __restrict__ bg2,  const float* __restrict__ Wq,
    const float* __restrict__ bq,   const float* __restrict__ Wk,
    const float* __restrict__ Wv,
    float* __restrict__ vg, float* __restrict__ weff, float* __restrict__ cq)
{
  __shared__ float A_lds[16][C_];    // 16 KB: zero-padded text_feat (A of GEMM1)
  __shared__ float T_lds[16][C2_];   // 32 KB: MLP hidden (rows 8..15 zero-padded)
  __shared__ float G_lds[B_][C_];    //  8 KB: GvT
  __shared__ float kg_lds[B_ * DK_]; //  1 KB

  const int tid  = threadIdx.x;
  const int wave = tid >> 5;
  const int lane = tid & 31;
  const int half = lane >> 4;   // K split for A/B fragments
  const int lmod = lane & 15;   // M (A) or N (B/C/D) within the 16-wide tile

  // stage zero-padded A once: unconditional ds_loads in the WMMA loop after this
  for (int idx = tid; idx < 16 * C_; idx += 256) {
    const int r = idx >> 8, c = idx & (C_ - 1);
    A_lds[r][c] = (r < B_) ? text[r * C_ + c] : 0.0f;
  }
  __syncthreads();

  // ---- GEMM1: T = pad16(text_feat)[16x256] @ Wg1[256x512] + bg1 ----
  for (int nt = wave; nt < C2_ / 16; nt += 8) {
    v8f acc = {};
    const int n = nt * 16 + lmod;
    for (int k0 = 0; k0 < C_; k0 += 4) {
      const int k = k0 + 2 * half;
      v2f a;  // A 16x4 f32: VGPR0 = K{k0 | k0+2}, VGPR1 = K{k0+1 | k0+3}
      a.x = A_lds[lmod][k];
      a.y = A_lds[lmod][k + 1];
      v2f b;  // B 4x16 f32, symmetric K striping
      b.x = Wg1[(k    ) * C2_ + n];
      b.y = Wg1[(k + 1) * C2_ + n];
      acc = __builtin_amdgcn_wmma_f32_16x16x4_f32(false, a, false, b,
                                                  (short)0, acc, false, false);
    }
#pragma unroll
    for (int r = 0; r < 8; ++r)                 // D: VGPR r -> M = r + 8*half
      T_lds[r + 8 * half][n] = acc[r] + bg1[n];
  }
  __syncthreads();

  // zero-pad rows 8..15 (A operand of GEMM2)
  for (int idx = tid; idx < 8 * C2_; idx += 256)
    T_lds[8 + (idx >> 9)][idx & (C2_ - 1)] = 0.0f;

  // ---- LayerNorm + exact GELU, one wave per valid row ----
  {
    const int r = wave;  // 0..7
    float s1 = 0.f, s2 = 0.f;
    for (int c = lane; c < C2_; c += 32) { float v = T_lds[r][c]; s1 += v; s2 += v * v; }
#pragma unroll
    for (int off = 16; off > 0; off >>= 1) {
      s1 += __shfl_xor(s1, off, 32);
      s2 += __shfl_xor(s2, off, 32);
    }
    const float mu  = s1 * (1.0f / C2_);
    const float var = s2 * (1.0f / C2_) - mu * mu;
    const float inv = rsqrtf(var + LN_EPS);
    for (int c = lane; c < C2_; c += 32) {
      float v = (T_lds[r][c] - mu) * inv * ln_g[c] + ln_b[c];
      T_lds[r][c] = 0.5f * v * (1.0f + erff(v * 0.70710678118f)); // exact GELU
    }
  }
  __syncthreads();

  // ---- GEMM2: GvT = T[16x512] @ Wg2[512x256] + bg2 ----
  for (int nt = wave; nt < C_ / 16; nt += 8) {
    v8f acc = {};
    const int n = nt * 16 + lmod;
    for (int k0 = 0; k0 < C2_; k0 += 4) {
      const int k = k0 + 2 * half;
      v2f a; a.x = T_lds[lmod][k]; a.y = T_lds[lmod][k + 1];
      v2f b; b.x = Wg2[(k) * C_ + n]; b.y = Wg2[(k + 1) * C_ + n];
      acc = __builtin_amdgcn_wmma_f32_16x16x4_f32(false, a, false, b,
                                                  (short)0, acc, false, false);
    }
#pragma unroll
    for (int r = 0; r < 8; ++r) {
      const int m = r + 8 * half;
      if (m < B_) G_lds[m][n] = acc[r] + bg2[n];
    }
  }
  __syncthreads();

  // ---- kg[b][d] = Wk[d,:] . GvT[b,:]  (one thread per output) ----
  {
    const int b = wave, d = lane;
    float acc = 0.f;
    for (int c = 0; c < C_; ++c) acc = fmaf(Wk[d * C_ + c], G_lds[b][c], acc);
    kg_lds[b * DK_ + d] = acc;
  }
  __syncthreads();

  // ---- vg[b][ch] = Wv[ch,:].GvT[b,:] ; weff[b][ch] = sum_d Wq[d,ch] kg[b,d] ----
  for (int idx = tid; idx < B_ * C_; idx += 256) {
    const int b = idx >> 8, ch = idx & (C_ - 1);
    float av = 0.f;
    for (int c = 0; c < C_; ++c) av = fmaf(Wv[ch * C_ + c], G_lds[b][c], av);
    vg[idx] = av;
    float w = 0.f;
#pragma unroll
    for (int d = 0; d < DK_; ++d) w = fmaf(Wq[d * C_ + ch], kg_lds[b * DK_ + d], w);
    weff[idx] = w;
  }
  if (tid < B_) {
    float acc = 0.f;
#pragma unroll
    for (int d = 0; d < DK_; ++d) acc = fmaf(bq[d], kg_lds[tid * DK_ + d], acc);
    cq[tid] = acc;
  }
}

// ---------------------------------------------------------------------------
// Kernel 2: a[b,i] = weff[b,:] . img[b,:,i] + cq[b]   (streams 19MB, coalesced)
// ---------------------------------------------------------------------------
__global__ __launch_bounds__(256) void tgca_proj_a(
    const float* __restrict__ img, const float* __restrict__ weff,
    const float* __restrict__ cq,  float* __restrict__ a_out)
{
  __shared__ float w_lds[C_];
  const int b = blockIdx.x / (HW_ / 256);
  const int i = (blockIdx.x % (HW_ / 256)) * 256 + threadIdx.x;
  w_lds[threadIdx.x] = weff[b * C_ + threadIdx.x];
  __syncthreads();
  const float* xb = img + (size_t)b * C_ * HW_ + i;
  float acc = cq[b];
#pragma unroll 4
  for (int c = 0; c < C_; ++c) {
    if ((c & 31) == 0 && c + 32 < C_)
      __builtin_prefetch(xb + (size_t)(c + 32) * HW_, 0, 0);  // global_prefetch_b8
    acc = fmaf(w_lds[c], xb[(size_t)c * HW_], acc);
  }
  a_out[b * HW_ + i] = acc;
}

// ---------------------------------------------------------------------------
// Kernel 3: s[b,i] = sum_j l_j exp(a*l_j - m) / sum_j exp(a*l_j - m)
// l is DMA'd into LDS via GLOBAL_LOAD_ASYNC_TO_LDS_B32 (ASYNCcnt), then the
// per-row softmax constants cancel; m = a>=0 ? a*max(l) : a*min(l).
// ---------------------------------------------------------------------------
__global__ __launch_bounds__(256) void tgca_softmax_collapse(
    const float* __restrict__ l, const float* __restrict__ a_in,
    float* __restrict__ s_out)
{
  __shared__ float l_lds[HW_];
  __shared__ float red_mx[8], red_mn[8];
  const int tid = threadIdx.x, wave = tid >> 5, lane = tid & 31;

  // Async copy l -> LDS. LDS byte address = low 32 bits of the shared-aperture
  // pointer (ISA §10.2).
  for (int j = tid; j < HW_; j += 256) {
    const unsigned int       lds_off = (unsigned int)(unsigned long long)(uintptr_t)&l_lds[j];
    const unsigned long long gaddr   = (unsigned long long)(uintptr_t)(l + j);
    asm volatile("global_load_async_to_lds_b32 %0, %1, off"
                 :: "v"(lds_off), "v"(gaddr) : "memory");
  }
  asm volatile("s_wait_asynccnt 0x0" ::: "memory");
  __syncthreads();

  float mx = -3.4e38f, mn = 3.4e38f;
  for (int j = tid; j < HW_; j += 256) {
    const float v = l_lds[j];
    mx = fmaxf(mx, v); mn = fminf(mn, v);
  }
#pragma unroll
  for (int off = 16; off > 0; off >>= 1) {
    mx = fmaxf(mx, __shfl_xor(mx, off, 32));
    mn = fminf(mn, __shfl_xor(mn, off, 32));
  }
  if (lane == 0) { red_mx[wave] = mx; red_mn[wave] = mn; }
  __syncthreads();
  float gmx = red_mx[0], gmn = red_mn[0];
#pragma unroll
  for (int w = 1; w < 8; ++w) { gmx = fmaxf(gmx, red_mx[w]); gmn = fminf(gmn, red_mn[w]); }

  const int pix = blockIdx.x * 256 + tid;
  const float a = a_in[pix];
  const float m = (a >= 0.f) ? a * gmx : a * gmn;
  float se = 0.f, sl = 0.f;
#pragma unroll 4
  for (int j = 0; j < HW_; ++j) {
    const float lj = l_lds[j];
    const float e  = __expf(fmaf(a, lj, -m));   // v_exp_f32 path
    se += e;
    sl  = fmaf(lj, e, sl);
  }
  s_out[pix] = sl / se;
}

// ---------------------------------------------------------------------------
// Kernel 4: out = img + gamma * (vg[b,c] * s[b,i] + bv[c])   (streams 38MB)
// ---------------------------------------------------------------------------
__global__ __launch_bounds__(256) void tgca_out(
    const float* __restrict__ img, const float* __restrict__ vg,
    const float* __restrict__ bv,  const float* __restrict__ s,
    const float* __restrict__ gamma, float* __restrict__ out)
{
  const size_t idx = (size_t)blockIdx.x * 256 + threadIdx.x;
  const int i  = (int)(idx % HW_);
  const int bc = (int)(idx / HW_);
  const int b  = bc >> 8;
  const int c  = bc & (C_ - 1);
  out[idx] = fmaf(gamma[0], fmaf(vg[bc], s[b * HW_ + i], bv[c]), img[idx]);
}

// ---------------------------------------------------------------------------
extern "C" void kernel_launch(void* const* d_in, const int* in_sizes, int n_in,
                              void* d_out, int out_size, void* d_ws, size_t ws_size,
                              hipStream_t stream) {
  const float* img   = (const float*)d_in[0];
  const float* text  = (const float*)d_in[1];
  const float* l     = (const float*)d_in[2];
  const float* Wg1   = (const float*)d_in[3];
  const float* bg1   = (const float*)d_in[4];
  const float* ln_g  = (const float*)d_in[5];
  const float* ln_b  = (const float*)d_in[6];
  const float* Wg2   = (const float*)d_in[7];
  const float* bg2   = (const float*)d_in[8];
  const float* Wq    = (const float*)d_in[9];
  const float* bq    = (const float*)d_in[10];
  const float* Wk    = (const float*)d_in[11];
  // d_in[12] = bk: cancels inside the softmax, provably unused.
  const float* Wv    = (const float*)d_in[13];
  const float* bv    = (const float*)d_in[14];
  const float* gamma = (const float*)d_in[15];
  float* out = (float*)d_out;

  // workspace (floats): vg[2048] | weff[2048] | cq[8 pad 256] | a[18432] | s[18432]
  float* ws   = (float*)d_ws;
  float* vg   = ws;
  float* weff = ws + 2048;
  float* cq   = ws + 4096;
  float* a    = ws + 4352;
  float* s    = ws + 4352 + B_ * HW_;

  hipLaunchKernelGGL(tgca_mlp_wmma, dim3(1), dim3(256), 0, stream,
                     text, Wg1, bg1, ln_g, ln_b, Wg2, bg2, Wq, bq, Wk, Wv,
                     vg, weff, cq);
  hipLaunchKernelGGL(tgca_proj_a, dim3(B_ * (HW_ / 256)), dim3(256), 0, stream,
                     img, weff, cq, a);
  hipLaunchKernelGGL(tgca_softmax_collapse, dim3((B_ * HW_) / 256), dim3(256), 0, stream,
                     l, a, s);
  hipLaunchKernelGGL(tgca_out, dim3((B_ * C_ * HW_) / 256), dim3(256), 0, stream,
                     img, vg, bv, s, gamma, out);
}